// ModelFilter_31860067402068
// MI455X (gfx1250) — compile-verified
//
#include <hip/hip_runtime.h>

typedef __attribute__((ext_vector_type(2))) float v2f;
typedef __attribute__((ext_vector_type(4))) float v4f;
typedef __attribute__((ext_vector_type(8))) float v8f;

#define T_LEN          262144
#define B_ROWS         64
#define CHUNK_L        256
#define CHUNKS_PER_ROW (T_LEN / CHUNK_L)      // 1024
#define WAVES_PER_ROW  (CHUNKS_PER_ROW / 16)  // 64
#define BLOCK_STEPS    (CHUNK_L / 16)         // 16
#define SQUARINGS      8                      // M^(2^8) = M^256

// workspace layout in floats
#define WS_HF    0                        // 16x16 row-major
#define WS_G1    256                      // 16
#define WS_G2    272                      // 16
#define WS_LOCAL 512                      // B*C*2 = 131072 floats
#define WS_INIT  (512 + (B_ROWS * CHUNKS_PER_ROW * 2))

__device__ __forceinline__ v8f wmma4(v2f a, v2f b, v8f c) {
    // D = A(16x4) * B(4x16) + C(16x16), f32
    return __builtin_amdgcn_wmma_f32_16x16x4_f32(
        /*neg_a=*/false, a, /*neg_b=*/false, b,
        /*c_mod=*/(short)0, c, /*reuse_a=*/false, /*reuse_b=*/false);
}

// --- 1-thread setup: build Hf (Toeplitz impulse response) and Hs columns ---
__global__ void iir_setup(const float* __restrict__ a_coeff, float* __restrict__ ws) {
    float a1 = a_coeff[1] / a_coeff[0];
    float a2 = a_coeff[2] / a_coeff[0];
    float h[16];
    h[0] = 1.0f; h[1] = -a1;
    for (int k = 2; k < 16; ++k) h[k] = -a1 * h[k - 1] - a2 * h[k - 2];
    for (int i = 0; i < 16; ++i)
        for (int j = 0; j < 16; ++j)
            ws[WS_HF + i * 16 + j] = (i >= j) ? h[i - j] : 0.0f;
    // homogeneous responses: g1 from y[-1]=1, g2 from y[-2]=1
    float g1m2 = 0.f, g1m1 = 1.f, g2m2 = 1.f, g2m1 = 0.f;
    for (int k = 0; k < 16; ++k) {
        float g1 = -a1 * g1m1 - a2 * g1m2;
        float g2 = -a1 * g2m1 - a2 * g2m2;
        ws[WS_G1 + k] = g1; ws[WS_G2 + k] = g2;
        g1m2 = g1m1; g1m1 = g1;
        g2m2 = g2m1; g2m1 = g2;
    }
}

// --- cross-chunk scan: entry state per chunk via M^CHUNK_L repeated squaring ---
__global__ void iir_combine(const float* __restrict__ a_coeff, float* __restrict__ ws) {
    int row = blockIdx.x * blockDim.x + threadIdx.x;
    if (row >= B_ROWS) return;
    double a1 = (double)a_coeff[1] / (double)a_coeff[0];
    double a2 = (double)a_coeff[2] / (double)a_coeff[0];
    double m00 = -a1, m01 = -a2, m10 = 1.0, m11 = 0.0;  // companion matrix
    for (int i = 0; i < SQUARINGS; ++i) {
        double t00 = m00 * m00 + m01 * m10;
        double t01 = m00 * m01 + m01 * m11;
        double t10 = m10 * m00 + m11 * m10;
        double t11 = m10 * m01 + m11 * m11;
        m00 = t00; m01 = t01; m10 = t10; m11 = t11;
    }
    float p00 = (float)m00, p01 = (float)m01, p10 = (float)m10, p11 = (float)m11;
    const float* ls = ws + WS_LOCAL + (long)row * CHUNKS_PER_ROW * 2;
    float* ini      = ws + WS_INIT  + (long)row * CHUNKS_PER_ROW * 2;
    float s0 = 0.f, s1 = 0.f;  // (y[-1], y[-2]) entering chunk
    for (int c = 0; c < CHUNKS_PER_ROW; ++c) {
        ini[2 * c]     = s0;
        ini[2 * c + 1] = s1;
        float n0 = p00 * s0 + p01 * s1 + ls[2 * c];
        float n1 = p10 * s0 + p11 * s1 + ls[2 * c + 1];
        s0 = n0; s1 = n1;
    }
}

// --- main pass: 1 wave = 16 chunks of one row; 16-sample steps via 5 WMMAs ---
template <bool WRITE_OUT>
__launch_bounds__(256)
__global__ void iir_pass(const float* __restrict__ x,
                         const float* __restrict__ a_coeff,
                         const float* __restrict__ b_coeff,
                         const float* __restrict__ gain,
                         float* __restrict__ ws,
                         float* __restrict__ out) {
    int tid  = blockIdx.x * blockDim.x + threadIdx.x;
    int w    = tid >> 5;
    int lane = tid & 31;
    int row   = w / WAVES_PER_ROW;
    int group = w % WAVES_PER_ROW;
    if (row >= B_ROWS) return;
    int col = lane & 15;   // N column == this lane's chunk within group
    int hi  = lane >> 4;   // 0: lanes 0-15, 1: lanes 16-31

    float inv_a0 = 1.0f / a_coeff[0];
    float b0 = b_coeff[0] * inv_a0;
    float b1 = b_coeff[1] * inv_a0;
    float b2 = b_coeff[2] * inv_a0;
    float g  = gain[0];

    long chunk = (long)group * 16 + col;
    const float* xrow = x + (long)row * T_LEN;
    long cb = chunk * CHUNK_L;

    // A matrices: Hf column slices. Lane l<16: row M=l holds K=4j,4j+1;
    // lane 16+l: row M=l holds K=4j+2,4j+3.
    v2f A[4];
    for (int j = 0; j < 4; ++j) {
        int c0 = 4 * j + hi * 2;
        A[j].x = ws[WS_HF + col * 16 + c0];
        A[j].y = ws[WS_HF + col * 16 + c0 + 1];
    }
    // State-injection A: columns (g1, g2, 0, 0)
    v2f As;
    As.x = hi ? 0.f : ws[WS_G1 + col];
    As.y = hi ? 0.f : ws[WS_G2 + col];

    // State-injection B: K-row 0 = y[-1] per chunk, K-row 1 = y[-2]; rows 2,3 = 0
    v2f Bs;
    if (WRITE_OUT) {
        const float* ini = ws + WS_INIT + ((long)row * CHUNKS_PER_ROW + chunk) * 2;
        Bs.x = hi ? 0.f : ini[0];
        Bs.y = hi ? 0.f : ini[1];
    } else {
        Bs.x = 0.f; Bs.y = 0.f;
    }

    for (int step = 0; step < BLOCK_STEPS; ++step) {
        int t0 = step * 16;
        // WGP-scope prefetch of next step's line (address always in-bounds:
        // guarded to stay inside this lane's chunk)
        if (step + 1 < BLOCK_STEPS)
            __builtin_prefetch(xrow + cb + t0 + 16 + hi * 2, 0, 3);

        // ---- FIR part first: independent of the carried state ----
        v8f acc = {};
        #pragma unroll
        for (int j = 0; j < 4; ++j) {
            int tb = t0 + 4 * j + hi * 2;      // even; this lane's two K-rows
            const float* p = xrow + cb + tb;
            float vx = p[0], vy = p[1];        // x[tb], x[tb+1]
            float ux, uy;                      // x[tb-2], x[tb-1]
            if (chunk == 0 && tb == 0) { ux = 0.f; uy = 0.f; }
            else { ux = p[-2]; uy = p[-1]; }
            v2f Bj;
            Bj.x = b0 * vx + b1 * uy + b2 * ux;   // f[tb]
            Bj.y = b0 * vy + b1 * vx + b2 * uy;   // f[tb+1]
            acc = wmma4(A[j], Bj, acc);
        }
        // ---- state injection LAST: the only op on the recurrent chain ----
        acc = wmma4(As, Bs, acc);

        // rows 15/14 of D live in lanes 16-31 (v7/v6); rotate into next Bs
        int idx = (col | 16) << 2;
        int i15 = __builtin_amdgcn_ds_bpermute(idx, __float_as_int(acc[7]));
        int i14 = __builtin_amdgcn_ds_bpermute(idx, __float_as_int(acc[6]));
        Bs.x = hi ? 0.f : __int_as_float(i15);
        Bs.y = hi ? 0.f : __int_as_float(i14);

        if (WRITE_OUT) {
            long base = (long)row * T_LEN + cb + t0 + hi * 8;
            v4f o0, o1;
            #pragma unroll
            for (int m = 0; m < 4; ++m) {
                float yv = fminf(1.0f, fmaxf(-1.0f, acc[m]));
                o0[m] = g * yv;
            }
            #pragma unroll
            for (int m = 0; m < 4; ++m) {
                float yv = fminf(1.0f, fmaxf(-1.0f, acc[4 + m]));
                o1[m] = g * yv;
            }
            *(v4f*)(out + base)     = o0;
            *(v4f*)(out + base + 4) = o1;
        }
    }

    if (!WRITE_OUT && !hi) {
        // final chunk-local state with zero ICs
        float* st = ws + WS_LOCAL + ((long)row * CHUNKS_PER_ROW + chunk) * 2;
        st[0] = Bs.x;   // y[L-1]
        st[1] = Bs.y;   // y[L-2]
    }
}

extern "C" void kernel_launch(void* const* d_in, const int* in_sizes, int n_in,
                              void* d_out, int out_size, void* d_ws, size_t ws_size,
                              hipStream_t stream) {
    const float* x       = (const float*)d_in[0];
    const float* a_coeff = (const float*)d_in[1];
    const float* b_coeff = (const float*)d_in[2];
    const float* gain    = (const float*)d_in[3];
    float* out = (float*)d_out;
    float* ws  = (float*)d_ws;

    iir_setup<<<1, 1, 0, stream>>>(a_coeff, ws);

    int total_threads = B_ROWS * WAVES_PER_ROW * 32;   // 4096 waves
    dim3 block(256);
    dim3 grid(total_threads / 256);

    iir_pass<false><<<grid, block, 0, stream>>>(x, a_coeff, b_coeff, gain, ws, out);
    iir_combine<<<1, 64, 0, stream>>>(a_coeff, ws);
    iir_pass<true><<<grid, block, 0, stream>>>(x, a_coeff, b_coeff, gain, ws, out);
}